// ConvTP_34531537060562
// MI455X (gfx1250) — compile-verified
//
#include <hip/hip_runtime.h>
#include <stdint.h>

#define IN_DIM 128
#define W_DIM 160          // 5*MUL floats per edge
#define OUT_DIM 224        // 7*MUL floats per node
#define INV_SQRT3 0.5773502691896258f
#define INV_SQRT2 0.7071067811865476f

#define WAVES_PER_BLOCK 8
#define EDGES_PER_STAGE 4
#define STAGE_FLOATS (EDGES_PER_STAGE * W_DIM)   // 640 dwords = 2560 B
#define STAGE_BYTES  (STAGE_FLOATS * 4)
#define CHUNKS 5          // fallback path: 5 x (32 lanes x 16 B) = 2560 B

typedef int      v4i  __attribute__((ext_vector_type(4)));
typedef unsigned u32x4 __attribute__((ext_vector_type(4)));
typedef int      i32x8 __attribute__((ext_vector_type(8)));
typedef int      i32x4 __attribute__((ext_vector_type(4)));
typedef __attribute__((address_space(1))) v4i gv4i;   // global int4
typedef __attribute__((address_space(3))) v4i lv4i;   // LDS int4
typedef __attribute__((address_space(3))) char lds_char;

#if __has_builtin(__builtin_amdgcn_tensor_load_to_lds)
#define USE_TDM 1
#else
#define USE_TDM 0
#endif

#if USE_TDM
// One TDM descriptor: 1D tile of 640 dwords (2560 B) global -> LDS.
__device__ __forceinline__ void tdm_copy_stage(unsigned lds_addr,
                                               unsigned long long gaddr) {
  u32x4 g0;
  g0.x = 1u;                                    // count=1 (valid), is_restore=0
  g0.y = lds_addr;                              // lds_addr[31:0]
  g0.z = (unsigned)gaddr;                       // global_addr[31:0]
  g0.w = (unsigned)((gaddr >> 32) & 0x1FFFFFFu) // global_addr[56:32]
         | (2u << 30);                          // type=2 ("image")
  i32x8 g1;
  g1.s0 = 0x20000;                              // data_size=2 (4 B), wg_mask=0
  g1.s1 = (STAGE_FLOATS & 0xFFFF) << 16;        // tensor_dim0[15:0]
  g1.s2 = 1 << 16;                              // tensor_dim0[31:16]=0, tensor_dim1=1
  g1.s3 = STAGE_FLOATS << 16;                   // tile_dim0=640
  g1.s4 = 0;                                    // tile_dim1=0 (unused), tile_dim2=0
  g1.s5 = STAGE_FLOATS;                         // tensor_dim0_stride low32
  g1.s6 = 0;
  g1.s7 = 0;
  i32x4 gz  = {0, 0, 0, 0};
  i32x8 gz8 = {0, 0, 0, 0, 0, 0, 0, 0};
  __builtin_amdgcn_tensor_load_to_lds(g0, g1, gz, gz, gz8, 0);
}
__device__ __forceinline__ void wait_stream_le1() {
  __builtin_amdgcn_s_wait_tensorcnt(1);
}
__device__ __forceinline__ void wait_stream_0() {
  __builtin_amdgcn_s_wait_tensorcnt(0);
}
#else
__device__ __forceinline__ void async_copy16(const void* g, void* l) {
#if __has_builtin(__builtin_amdgcn_global_load_async_to_lds_b128)
  __builtin_amdgcn_global_load_async_to_lds_b128((gv4i*)g, (lv4i*)l, 0, 0);
#else
  unsigned lds_addr = (unsigned)(uintptr_t)(lds_char*)l;
  unsigned long long gaddr = (unsigned long long)(uintptr_t)g;
  asm volatile("global_load_async_to_lds_b128 %0, %1, off"
               :: "v"(lds_addr), "v"(gaddr) : "memory");
#endif
}
__device__ __forceinline__ void wait_stream_le1() {
#if __has_builtin(__builtin_amdgcn_s_wait_asynccnt)
  __builtin_amdgcn_s_wait_asynccnt(CHUNKS);
#else
  asm volatile("s_wait_asynccnt 0x5" ::: "memory");
#endif
}
__device__ __forceinline__ void wait_stream_0() {
#if __has_builtin(__builtin_amdgcn_s_wait_asynccnt)
  __builtin_amdgcn_s_wait_asynccnt(0);
#else
  asm volatile("s_wait_asynccnt 0x0" ::: "memory");
#endif
}
#endif

__global__ __launch_bounds__(256) void ConvTP_kernel(
    const float* __restrict__ nf,        // [nNodes, 128]
    const float* __restrict__ ang,       // [nEdges, 4]
    const long long* __restrict__ eidx,  // [nEdges, 2] int64
    const float* __restrict__ tpw,       // [nEdges, 160]
    float* __restrict__ out,             // [nNodes, 224] pre-zeroed
    int nEdges, int edgesPerWave)
{
  __shared__ __align__(16) float smem[2][WAVES_PER_BLOCK][STAGE_FLOATS]; // 40 KB

  const int tid  = (int)threadIdx.x;
  const int lane = tid & 31;
  const int wid  = tid >> 5;

  const int wgid = (int)blockIdx.x * WAVES_PER_BLOCK + wid;
  const int e0 = wgid * edgesPerWave;
  int e1 = e0 + edgesPerWave;
  if (e1 > nEdges) e1 = nEdges;
  if (e0 >= e1) return;
  const int nStages = (e1 - e0 + EDGES_PER_STAGE - 1) / EDGES_PER_STAGE;

  // Total weight bytes = nEdges*640 < 2^31 -> 32-bit offsets are safe.
  const unsigned tpw_end = (unsigned)nEdges * (unsigned)(W_DIM * 4);
  const unsigned long long tpw_base = (unsigned long long)(uintptr_t)tpw;

  // Stage 4 edges' weights (2560 B) into this wave's LDS slice.
  auto stage = [&](int s, int buf) {
    unsigned goff = (unsigned)(e0 + s * EDGES_PER_STAGE) * (unsigned)(W_DIM * 4);
    if (goff + STAGE_BYTES > tpw_end) goff = 0;  // clamp OOB tail (data unused)
#if USE_TDM
    unsigned lds_addr =
        (unsigned)(uintptr_t)(lds_char*)&smem[buf][wid][0];
    tdm_copy_stage(lds_addr, tpw_base + goff);
#else
    const char* gbase = (const char*)tpw + goff + lane * 16;
    char* lbase = (char*)&smem[buf][wid][0] + lane * 16;
#pragma unroll
    for (int i = 0; i < CHUNKS; ++i)
      async_copy16(gbase + i * 512, lbase + i * 512);
#endif
  };

  stage(0, 0);

  for (int s = 0; s < nStages; ++s) {
    if (s + 1 < nStages) {
      stage(s + 1, (s + 1) & 1);
      wait_stream_le1();   // newest stage may be in flight -> buffer s complete
    } else {
      wait_stream_0();
    }

    const float* wS = &smem[s & 1][wid][0];
    const int eBase = e0 + s * EDGES_PER_STAGE;
    const int rem = e1 - eBase;
    const int cnt = rem < EDGES_PER_STAGE ? rem : EDGES_PER_STAGE;

    for (int le = 0; le < cnt; ++le) {
      const int e = eBase + le;
      // wave-uniform per-edge scalars (broadcast loads, L2-resident)
      const int src = (int)eidx[2 * e];
      const int dst = (int)eidx[2 * e + 1];
      const float y0  = ang[4 * e + 0];
      const float y1a = ang[4 * e + 1];
      const float y1b = ang[4 * e + 2];
      const float y1c = ang[4 * e + 3];

      // coalesced gather of the source node row (lane = channel)
      const float* hrow = nf + src * IN_DIM;
      const float h0  = hrow[lane];
      const float h1a = hrow[32 + lane];
      const float h1b = hrow[64 + lane];
      const float h1c = hrow[96 + lane];

      // weights from LDS (bank-conflict-free: stride 32 dwords)
      const float* w = wS + le * W_DIM;
      const float w0 = w[lane];
      const float w1 = w[32 + lane];
      const float w2 = w[64 + lane];
      const float w3 = w[96 + lane];
      const float w4 = w[128 + lane];

      const float dot = h1a * y1a + h1b * y1b + h1c * y1c;
      const float out0 = w0 * h0 * y0 + w3 * (INV_SQRT3 * dot);

      const float o1o_a = w1 * h0 * y1a + w2 * h1a * y0;
      const float o1o_b = w1 * h0 * y1b + w2 * h1b * y0;
      const float o1o_c = w1 * h0 * y1c + w2 * h1c * y0;

      const float cx = h1b * y1c - h1c * y1b;
      const float cy = h1c * y1a - h1a * y1c;
      const float cz = h1a * y1b - h1b * y1a;
      const float k4 = w4 * INV_SQRT2;

      // segment-sum scatter: non-returning f32 atomics, resolve in L2
      float* orow = out + dst * OUT_DIM;
      atomicAdd(orow + lane,        out0);
      atomicAdd(orow +  32 + lane,  o1o_a);
      atomicAdd(orow +  64 + lane,  o1o_b);
      atomicAdd(orow +  96 + lane,  o1o_c);
      atomicAdd(orow + 128 + lane,  k4 * cx);
      atomicAdd(orow + 160 + lane,  k4 * cy);
      atomicAdd(orow + 192 + lane,  k4 * cz);
    }
  }
}

extern "C" void kernel_launch(void* const* d_in, const int* in_sizes, int n_in,
                              void* d_out, int out_size, void* d_ws, size_t ws_size,
                              hipStream_t stream) {
  (void)n_in; (void)d_ws; (void)ws_size;
  const float*     nf   = (const float*)d_in[0];
  const float*     ang  = (const float*)d_in[1];
  const long long* eidx = (const long long*)d_in[2];
  const float*     tpw  = (const float*)d_in[3];
  float*           out  = (float*)d_out;

  const int nEdges = in_sizes[2] / 2;

  // output is accumulated via atomics -> must start at zero every call
  (void)hipMemsetAsync(d_out, 0, (size_t)out_size * sizeof(float), stream);

  const int blocks = 1024;                       // 8192 waves
  const long long totalWaves = (long long)blocks * WAVES_PER_BLOCK;
  const int edgesPerWave = (int)((nEdges + totalWaves - 1) / totalWaves);

  ConvTP_kernel<<<blocks, 256, 0, stream>>>(nf, ang, eidx, tpw, out,
                                            nEdges, edgesPerWave);
}